// MMBasicLayer_53987738911561
// MI455X (gfx1250) — compile-verified
//
#include <hip/hip_runtime.h>
#include <hip/hip_bf16.h>
#include <math.h>

// ---------------------------------------------------------------------------
// MI455X (gfx1250) fused PWAM. All heavy GEMMs via V_WMMA_F32_16X16X4_F32
// (wave32, full f32 precision). InstanceNorms algebraically folded into
// adjacent GEMM weights/epilogues; stats gathered via f32 global atomics.
// Workspace requirement: ~556 MB (large L2/HBM; all GEMM operands L2-resident).
// ---------------------------------------------------------------------------

typedef float v2f __attribute__((ext_vector_type(2)));
typedef float v8f __attribute__((ext_vector_type(8)));

#define HW   16384
#define NB   8
#define NFAM 896   // per-batch stat channels: q1@0(128) kl@128(256) vv@384(256) q2@640(128) tW@768(128)

// workspace offsets (in floats)
static constexpr size_t OFF_WBIG   = 0;          //  768*128
static constexpr size_t OFF_BIAS   = 98304;      //  768
static constexpr size_t OFF_SUMS   = 99072;      //  8*896*2
static constexpr size_t OFF_MURS   = 113408;     //  8*896*2
static constexpr size_t OFF_WFOLD  = 127744;     //  8*128*128
static constexpr size_t OFF_BFOLD  = 258816;     //  8*128
static constexpr size_t OFF_KFOLD  = 259840;     //  8*128*20
static constexpr size_t OFF_WV     = 280320;     //  8*128*20
static constexpr size_t OFF_LBIAS  = 300800;     //  8*20
static constexpr size_t OFF_QLF    = 300960;     //  8*256*20
static constexpr size_t OFF_LCONST = 341920;     //  8*20
static constexpr size_t OFF_OUTPRE = 342080;     //  8*256*20
static constexpr size_t OFF_Y1     = 393216;     //  8*16384*768  (vis|q1|kl|vv)
static constexpr size_t OFF_TQ2    = 101056512;  //  8*16384*128
static constexpr size_t OFF_TW     = 117833728;  //  8*16384*128
static constexpr size_t OFF_L      = 134610944;  //  8*32*16384   (sim_l logits -> probs, rows 20..31 zero)

__device__ __forceinline__ float gelu_erf(float x) {
  return 0.5f * x * (1.0f + erff(x * 0.70710678118654752f));
}

__device__ __forceinline__ v8f wmma4(v2f a, v2f b, v8f c) {
  // D = A(16x4,f32) * B(4x16,f32) + C(16x16,f32)
  return __builtin_amdgcn_wmma_f32_16x16x4_f32(false, a, false, b, (short)0, c, false, false);
}

// ---------------------------------------------------------------- utilities
__global__ void zero_f(float* __restrict__ p, int n) {
  int i = blockIdx.x * 256 + threadIdx.x;
  if (i < n) p[i] = 0.f;
}

__global__ void pack_w(const float* __restrict__ visW, const float* __restrict__ visB,
                       const float* __restrict__ rvfqW, const float* __restrict__ rvfqB,
                       const float* __restrict__ rlfkW, const float* __restrict__ rlfkB,
                       const float* __restrict__ fvvW, const float* __restrict__ fvvB,
                       float* __restrict__ Wbig, float* __restrict__ bias) {
  int idx = blockIdx.x * 256 + threadIdx.x;
  if (idx < 98304) {
    int r = idx >> 7, i = idx & 127;
    float v;
    if (r < 128)      v = visW[r * 128 + i];
    else if (r < 256) v = rvfqW[(r - 128) * 128 + i];
    else if (r < 512) v = rlfkW[(r - 256) * 128 + i];
    else              v = fvvW[(r - 512) * 128 + i];
    Wbig[idx] = v;
  } else if (idx < 99072) {
    int ch = idx - 98304;
    float v;
    if (ch < 128)      v = visB[ch];
    else if (ch < 256) v = rvfqB[ch - 128];
    else if (ch < 512) v = rlfkB[ch - 256];
    else               v = fvvB[ch - 512];
    bias[ch] = v;
  }
}

// ------------------------------------------------ K1: Wbig(768x128) @ x^T, per batch
// Y1[b][n][0:128]=gelu(vis), [128:256]=t_q1, [256:512]=t_kl, [512:768]=t_vv
__global__ __launch_bounds__(256) void k1_gemm(const float* __restrict__ x,
                                               const float* __restrict__ Wbig,
                                               const float* __restrict__ bias768,
                                               float* __restrict__ Y1) {
  int wid  = blockIdx.x * 8 + (threadIdx.x >> 5);
  int lane = threadIdx.x & 31;
  int b  = wid / (48 * 256);
  int r  = wid % (48 * 256);
  int mt = r >> 8;                    // 0..47 (16-row M tile)
  int ng = r & 255;                   // 64-col N group
  int koff = (lane >> 4) << 1;
  const float* Arow = Wbig + (size_t)(mt * 16 + (lane & 15)) * 128 + koff;
  size_t xbase = ((size_t)b * HW + (size_t)ng * 64 + (lane & 15)) * 128 + koff;
  v8f acc0 = {0,0,0,0,0,0,0,0}, acc1 = acc0, acc2 = acc0, acc3 = acc0;
  for (int k0 = 0; k0 < 128; k0 += 4) {
    __builtin_prefetch((const void*)(x + xbase + k0 + 64), 0, 0);  // global_prefetch_b8
    v2f a  = *(const v2f*)(Arow + k0);
    v2f b0 = *(const v2f*)(x + xbase + 0 * 16 * 128 + k0);
    v2f b1 = *(const v2f*)(x + xbase + 1 * 16 * 128 + k0);
    v2f b2 = *(const v2f*)(x + xbase + 2 * 16 * 128 + k0);
    v2f b3 = *(const v2f*)(x + xbase + 3 * 16 * 128 + k0);
    acc0 = wmma4(a, b0, acc0);
    acc1 = wmma4(a, b1, acc1);
    acc2 = wmma4(a, b2, acc2);
    acc3 = wmma4(a, b3, acc3);
  }
  int ch0 = mt * 16 + ((lane >> 4) << 3);  // 8 consecutive output channels per lane
  float bia[8];
#pragma unroll
  for (int v = 0; v < 8; ++v) bia[v] = bias768[ch0 + v];
  bool is_vis = (mt < 8);
  v8f accs[4] = {acc0, acc1, acc2, acc3};
#pragma unroll
  for (int s = 0; s < 4; ++s) {
    int n = ng * 64 + s * 16 + (lane & 15);
    float o[8];
#pragma unroll
    for (int v = 0; v < 8; ++v) {
      float t = accs[s][v] + bia[v];
      o[v] = is_vis ? gelu_erf(t) : t;
    }
    float* dst = Y1 + ((size_t)b * HW + n) * 768 + ch0;
    *(float4*)(dst)     = make_float4(o[0], o[1], o[2], o[3]);
    *(float4*)(dst + 4) = make_float4(o[4], o[5], o[6], o[7]);
  }
}

// ----------------------------------------- stats: sum & sumsq per (b, channel)
__global__ __launch_bounds__(256) void stats_sum(const float* __restrict__ src,
                                                 int ld, int cofs, int famoff,
                                                 float* __restrict__ sums) {
  __shared__ float2 red[256];
  int b = blockIdx.z, g = blockIdx.y;
  int c = threadIdx.x & 63, nsub = threadIdx.x >> 6;
  float s0 = 0.f, s1 = 0.f;
  size_t base = ((size_t)b * HW + blockIdx.x * 256 + nsub * 64) * (size_t)ld + cofs + g * 64 + c;
  for (int i = 0; i < 64; ++i) {
    float v = src[base + (size_t)i * ld];
    s0 += v;
    s1 = fmaf(v, v, s1);
  }
  red[threadIdx.x] = make_float2(s0, s1);
  __syncthreads();
  if (threadIdx.x < 64) {
    float2 a = red[threadIdx.x], bb = red[threadIdx.x + 64];
    float2 cc = red[threadIdx.x + 128], dd = red[threadIdx.x + 192];
    float t0 = a.x + bb.x + cc.x + dd.x;
    float t1 = a.y + bb.y + cc.y + dd.y;
    size_t o = ((size_t)b * NFAM + famoff + g * 64 + c) * 2;
    atomicAdd(&sums[o], t0);
    atomicAdd(&sums[o + 1], t1);
  }
}

__global__ void stats_fin(const float* __restrict__ sums, float* __restrict__ murs,
                          int famoff, int count) {
  int idx = blockIdx.x * 256 + threadIdx.x;
  if (idx >= NB * count) return;
  int b = idx / count, k = idx % count;
  size_t o = ((size_t)b * NFAM + famoff + k) * 2;
  float mu  = sums[o] * (1.0f / HW);
  float var = fmaxf(sums[o + 1] * (1.0f / HW) - mu * mu, 0.f);
  murs[o]     = mu;
  murs[o + 1] = rsqrtf(var + 1e-5f);
}

// -------------------------- fold inorm(q1) into rv_fq2 weights (per batch)
__global__ void k2b_wfold(const float* __restrict__ rvfq2W, const float* __restrict__ rvfq2B,
                          const float* __restrict__ murs,
                          float* __restrict__ Wfold, float* __restrict__ bfold) {
  int b = blockIdx.x, o = threadIdx.x;  // 128 threads
  float dot = 0.f;
  for (int i = 0; i < 128; ++i) {
    float mu = murs[((size_t)b * NFAM + i) * 2];
    float rs = murs[((size_t)b * NFAM + i) * 2 + 1];
    float w  = rvfq2W[o * 128 + i] * rs;
    Wfold[((size_t)b * 128 + o) * 128 + i] = w;
    dot = fmaf(w, mu, dot);
  }
  bfold[b * 128 + o] = rvfq2B[o] - dot;
}

// ------------------------------------- K3: t_q2 = Wfold[b] @ t_q1 + bfold[b]
__global__ __launch_bounds__(256) void k3_gemm(const float* __restrict__ Y1,
                                               const float* __restrict__ Wfold,
                                               const float* __restrict__ bfold,
                                               float* __restrict__ tq2) {
  int wid  = blockIdx.x * 8 + (threadIdx.x >> 5);
  int lane = threadIdx.x & 31;
  int b  = wid >> 11;
  int r  = wid & 2047;
  int mt = r >> 8;   // 0..7
  int ng = r & 255;
  int koff = (lane >> 4) << 1;
  const float* Arow = Wfold + (size_t)b * 128 * 128 + (size_t)(mt * 16 + (lane & 15)) * 128 + koff;
  size_t bbase = ((size_t)b * HW + (size_t)ng * 64 + (lane & 15)) * 768 + 128 + koff;
  v8f acc0 = {0,0,0,0,0,0,0,0}, acc1 = acc0, acc2 = acc0, acc3 = acc0;
  for (int k0 = 0; k0 < 128; k0 += 4) {
    v2f a  = *(const v2f*)(Arow + k0);
    v2f b0 = *(const v2f*)(Y1 + bbase + 0 * 16 * 768 + k0);
    v2f b1 = *(const v2f*)(Y1 + bbase + 1 * 16 * 768 + k0);
    v2f b2 = *(const v2f*)(Y1 + bbase + 2 * 16 * 768 + k0);
    v2f b3 = *(const v2f*)(Y1 + bbase + 3 * 16 * 768 + k0);
    acc0 = wmma4(a, b0, acc0);
    acc1 = wmma4(a, b1, acc1);
    acc2 = wmma4(a, b2, acc2);
    acc3 = wmma4(a, b3, acc3);
  }
  int ch0 = mt * 16 + ((lane >> 4) << 3);
  float bi[8];
#pragma unroll
  for (int v = 0; v < 8; ++v) bi[v] = bfold[b * 128 + ch0 + v];
  v8f accs[4] = {acc0, acc1, acc2, acc3};
#pragma unroll
  for (int s = 0; s < 4; ++s) {
    int n = ng * 64 + s * 16 + (lane & 15);
    float* dst = tq2 + ((size_t)b * HW + n) * 128 + ch0;
    *(float4*)(dst)     = make_float4(accs[s][0] + bi[0], accs[s][1] + bi[1],
                                      accs[s][2] + bi[2], accs[s][3] + bi[3]);
    *(float4*)(dst + 4) = make_float4(accs[s][4] + bi[4], accs[s][5] + bi[5],
                                      accs[s][6] + bi[6], accs[s][7] + bi[7]);
  }
}

// ------------- per-batch small tensors: kfold/lbias (vis-attn), Wv, qlf/lconst
__global__ __launch_bounds__(256) void small_pre(
    const float* __restrict__ l, const float* __restrict__ lm,
    const float* __restrict__ rvfkW, const float* __restrict__ rvfkB,
    const float* __restrict__ fvW, const float* __restrict__ fvB,
    const float* __restrict__ WW,
    const float* __restrict__ rlfqW, const float* __restrict__ rlfqB,
    const float* __restrict__ rlfq2W, const float* __restrict__ rlfq2B,
    const float* __restrict__ murs,
    float* __restrict__ kfold, float* __restrict__ Wv,
    float* __restrict__ lbias, float* __restrict__ qlf, float* __restrict__ lconst) {
  __shared__ float s_l[5120];     // l (256x20); reused for ql2 (256x20) in phase 2
  __shared__ float s_kraw[2560];
  __shared__ float s_val[2560];
  __shared__ float s_ql[2560];
  __shared__ float s_mask[20];
  __shared__ float s_mu2[128], s_rs2[128];
  __shared__ float s_mukl[256], s_rskl[256];
  int b = blockIdx.x, t = threadIdx.x;
  for (int i = t; i < 5120; i += 256) s_l[i] = l[(size_t)b * 5120 + i];
  if (t < 20) s_mask[t] = lm[b * 20 + t];
  if (t < 128) {
    s_mu2[t] = murs[((size_t)b * NFAM + 640 + t) * 2];
    s_rs2[t] = murs[((size_t)b * NFAM + 640 + t) * 2 + 1];
  }
  for (int i = t; i < 256; i += 256) {
    s_mukl[i] = murs[((size_t)b * NFAM + 128 + i) * 2];
    s_rskl[i] = murs[((size_t)b * NFAM + 128 + i) * 2 + 1];
  }
  __syncthreads();
  const float scale = 0.08838834764831845f;  // 128^-0.5
  // phase 1: k_raw, value, ql (each 128x20)
  for (int idx = t; idx < 2560; idx += 256) {
    int c = idx / 20, j = idx % 20;
    float dk = rvfkB[c], dv = fvB[c], dq = rlfqB[c];
    for (int i = 0; i < 256; ++i) {
      float lv = s_l[i * 20 + j];
      dk = fmaf(rvfkW[c * 256 + i], lv, dk);
      dv = fmaf(fvW[c * 256 + i], lv, dv);
      dq = fmaf(rlfqW[c * 256 + i], lv, dq);
    }
    float mj = s_mask[j];
    dk *= mj; dv *= mj;
    s_kraw[idx] = dk; s_val[idx] = dv; s_ql[idx] = dq;
    kfold[(size_t)b * 2560 + idx] = dk * s_rs2[c] * scale;  // q2-norm folded into k
  }
  __syncthreads();
  // phase 2a: lbias_j = -scale*sum_c mu2*rs2*k_raw + key-mask bias
  if (t < 20) {
    float s = 0.f;
    for (int c = 0; c < 128; ++c) s = fmaf(s_mu2[c] * s_rs2[c], s_kraw[c * 20 + t], s);
    lbias[b * 20 + t] = -scale * s + (10000.f * s_mask[t] - 10000.f);
  }
  // phase 2b: ql2 = rl_fq2 @ ql (masked); fold kl-norm scale -> qlf
  for (int idx = t; idx < 5120; idx += 256) {
    int c2 = idx / 20, j = idx % 20;
    float d = rlfq2B[c2];
    for (int i = 0; i < 128; ++i) d = fmaf(rlfq2W[c2 * 128 + i], s_ql[i * 20 + j], d);
    d *= s_mask[j];
    s_l[idx] = d;  // reuse as ql2
    qlf[(size_t)b * 5120 + idx] = d * s_rskl[c2] * scale;
  }
  // phase 2c: Wv = W_W @ value  (W-conv folded through the attention)
  for (int idx = t; idx < 2560; idx += 256) {
    int c = idx / 20, j = idx % 20;
    float d = 0.f;
    for (int i = 0; i < 128; ++i) d = fmaf(WW[c * 128 + i], s_val[i * 20 + j], d);
    Wv[(size_t)b * 2560 + idx] = d;
  }
  __syncthreads();
  // phase 3: lconst_j = -scale*sum_c ql2*rskl*mukl
  if (t < 20) {
    float s = 0.f;
    for (int c2 = 0; c2 < 256; ++c2) s = fmaf(s_l[c2 * 20 + t] * s_rskl[c2], s_mukl[c2], s);
    lconst[b * 20 + t] = -scale * s;
  }
}

// --------------------- K6: per-position visual attention -> t_W (pre-inorm)
__global__ __launch_bounds__(256) void k6_attn(const float* __restrict__ tq2,
                                               const float* __restrict__ kfold,
                                               const float* __restrict__ Wv,
                                               const float* __restrict__ lbias,
                                               const float* __restrict__ Wb,
                                               float* __restrict__ tW) {
  __shared__ float s_kf[2560], s_wv[2560], s_lb[20], s_wb[128];
  int b = blockIdx.x >> 6, chunk = blockIdx.x & 63;
  for (int i = threadIdx.x; i < 2560; i += 256) {
    s_kf[i] = kfold[(size_t)b * 2560 + i];
    s_wv[i] = Wv[(size_t)b * 2560 + i];
  }
  if (threadIdx.x < 20) s_lb[threadIdx.x] = lbias[b * 20 + threadIdx.x];
  if (threadIdx.x < 128) s_wb[threadIdx.x] = Wb[threadIdx.x];
  __syncthreads();
  int n = chunk * 256 + threadIdx.x;
  const float* q = tq2 + ((size_t)b * HW + n) * 128;
  float lg[20];
#pragma unroll
  for (int j = 0; j < 20; ++j) lg[j] = s_lb[j];
  for (int c = 0; c < 128; ++c) {
    float qc = q[c];
#pragma unroll
    for (int j = 0; j < 20; ++j) lg[j] = fmaf(qc, s_kf[c * 20 + j], lg[j]);
  }
  float mx = lg[0];
#pragma unroll
  for (int j = 1; j < 20; ++j) mx = fmaxf(mx, lg[j]);
  float sum = 0.f;
#pragma unroll
  for (int j = 0; j < 20; ++j) { lg[j] = expf(lg[j] - mx); sum += lg[j]; }
  float inv = 1.f / sum;
#pragma unroll
  for (int j = 0; j < 20; ++j) lg[j] *= inv;
  float* dst = tW + ((size_t)b * HW + n) * 128;
  for (int c = 0; c < 128; c += 4) {
    float4 o;
    float t0 = s_wb[c], t1 = s_wb[c + 1], t2 = s_wb[c + 2], t3 = s_wb[c + 3];
#pragma unroll
    for (int j = 0; j < 20; ++j) {
      float p = lg[j];
      t0 = fmaf(p, s_wv[(c + 0) * 20 + j], t0);
      t1 = fmaf(p, s_wv[(c + 1) * 20 + j], t1);
      t2 = fmaf(p, s_wv[(c + 2) * 20 + j], t2);
      t3 = fmaf(p, s_wv[(c + 3) * 20 + j], t3);
    }
    o.x = t0; o.y = t1; o.z = t2; o.w = t3;
    *(float4*)(dst + c) = o;
  }
}

// ---- K8: mm = gelu(pm_W @ (vis ⊙ inorm(t_W)) + pm_b), stored (B,HW,C) -> d_out
__global__ __launch_bounds__(256) void k8_mm(const float* __restrict__ Y1,
                                             const float* __restrict__ tW,
                                             const float* __restrict__ murs,
                                             const float* __restrict__ pmW,
                                             const float* __restrict__ pmB,
                                             float* __restrict__ out) {
  __shared__ float s_mu[128], s_rs[128];
  int b = blockIdx.x >> 8;
  if (threadIdx.x < 128) {
    s_mu[threadIdx.x] = murs[((size_t)b * NFAM + 768 + threadIdx.x) * 2];
    s_rs[threadIdx.x] = murs[((size_t)b * NFAM + 768 + threadIdx.x) * 2 + 1];
  }
  __syncthreads();
  int lw   = (blockIdx.x & 255) * 8 + (threadIdx.x >> 5);
  int lane = threadIdx.x & 31;
  int mt = lw >> 8;   // 0..7
  int ng = lw & 255;
  int koff = (lane >> 4) << 1;
  const float* Arow = pmW + (size_t)(mt * 16 + (lane & 15)) * 128 + koff;
  size_t visbase = ((size_t)b * HW + (size_t)ng * 64 + (lane & 15)) * 768 + koff;
  size_t twbase  = ((size_t)b * HW + (size_t)ng * 64 + (lane & 15)) * 128 + koff;
  v8f acc0 = {0,0,0,0,0,0,0,0}, acc1 = acc0, acc2 = acc0, acc3 = acc0;
  for (int k0 = 0; k0 < 128; k0 += 4) {
    v2f a = *(const v2f*)(Arow + k0);
    float mu0 = s_mu[k0 + koff], mu1 = s_mu[k0 + koff + 1];
    float rs0 = s_rs[k0 + koff], rs1 = s_rs[k0 + koff + 1];
#pragma unroll
    for (int s = 0; s < 4; ++s) {
      v2f vis = *(const v2f*)(Y1 + visbase + (size_t)s * 16 * 768 + k0);
      v2f tw  = *(const v2f*)(tW + twbase + (size_t)s * 16 * 128 + k0);
      v2f h;
      h.x = vis.x * ((tw.x - mu0) * rs0);
      h.y = vis.y * ((tw.y - mu1) * rs1);
      if (s == 0) acc0 = wmma4(a, h, acc0);
      else if (s == 1) acc1 = wmma4(a, h, acc1);
      else if (s == 2) acc2 = wmma4(a, h, acc2);
      else acc3 = wmma4(a, h, acc3);
    }
  }
  int ch0 = mt * 16 + ((lane >> 4) << 3);
  float bi[8];
#pragma unroll
  for (int v = 0; v < 8; ++v) bi[v] = pmB[ch0 + v];
  v8f accs[4] = {acc0, acc1, acc2, acc3};
#pragma unroll
  for (int s = 0; s < 4; ++s) {
    int n = ng * 64 + s * 16 + (lane & 15);
    float o[8];
#pragma unroll
    for (int v = 0; v < 8; ++v) o[v] = gelu_erf(accs[s][v] + bi[v]);
    float* dst = out + ((size_t)b * HW + n) * 128 + ch0;
    *(float4*)(dst)     = make_float4(o[0], o[1], o[2], o[3]);
    *(float4*)(dst + 4) = make_float4(o[4], o[5], o[6], o[7]);
  }
}

// ------------------------- lang1: sim_l logits[b][j][h] (kl-norm folded in qlf)
__global__ __launch_bounds__(256) void lang1(const float* __restrict__ Y1,
                                             const float* __restrict__ qlf,
                                             const float* __restrict__ lconst,
                                             float* __restrict__ L) {
  __shared__ float s_q[5120], s_c[20];
  int b = blockIdx.x >> 6, chunk = blockIdx.x & 63;
  for (int i = threadIdx.x; i < 5120; i += 256) s_q[i] = qlf[(size_t)b * 5120 + i];
  if (threadIdx.x < 20) s_c[threadIdx.x] = lconst[b * 20 + threadIdx.x];
  __syncthreads();
  int h = chunk * 256 + threadIdx.x;
  const float* kl = Y1 + ((size_t)b * HW + h) * 768 + 256;
  float lg[20];
#pragma unroll
  for (int j = 0; j < 20; ++j) lg[j] = s_c[j];
  for (int c = 0; c < 256; ++c) {
    float v = kl[c];
#pragma unroll
    for (int j = 0; j < 20; ++j) lg[j] = fmaf(v, s_q[c * 20 + j], lg[j]);
  }
#pragma unroll
  for (int j = 0; j < 20; ++j) L[((size_t)(b * 32 + j)) * HW + h] = lg[j];
}

// ------------ lang2: rowwise softmax over HW; rows 20..31 zeroed (WMMA padding)
__global__ __launch_bounds__(256) void lang2(float* __restrict__ L) {
  __shared__ float red[256];
  int row = blockIdx.x;           // b*32 + j
  float* p = L + (size_t)row * HW;
  if ((row & 31) >= 20) {
    for (int i = threadIdx.x; i < HW; i += 256) p[i] = 0.f;
    return;
  }
  float mx = -3.4e38f;
  for (int i = threadIdx.x; i < HW; i += 256) mx = fmaxf(mx, p[i]);
  red[threadIdx.x] = mx;
  __syncthreads();
  for (int s = 128; s > 0; s >>= 1) {
    if (threadIdx.x < s) red[threadIdx.x] = fmaxf(red[threadIdx.x], red[threadIdx.x + s]);
    __syncthreads();
  }
  mx = red[0];
  __syncthreads();
  float sum = 0.f;
  for (int i = threadIdx.x; i < HW; i += 256) {
    float e = expf(p[i] - mx);
    p[i] = e;
    sum += e;
  }
  red[threadIdx.x] = sum;
  __syncthreads();
  for (int s = 128; s > 0; s >>= 1) {
    if (threadIdx.x < s) red[threadIdx.x] += red[threadIdx.x + s];
    __syncthreads();
  }
  float inv = 1.f / red[0];
  for (int i = threadIdx.x; i < HW; i += 256) p[i] *= inv;
}

// ---- lang3: out_pre = inorm(vv) @ P^T via WMMA (K=16384); vv-norm folded in
//      epilogue using sum(P row)==1.
__global__ __launch_bounds__(256) void lang3_gemm(const float* __restrict__ Y1,
                                                  const float* __restrict__ L,
                                                  const float* __restrict__ murs,
                                                  float* __restrict__ outpre) {
  int wid  = blockIdx.x * 8 + (threadIdx.x >> 5);
  int lane = threadIdx.x & 31;
  int b  = wid >> 5;
  int r  = wid & 31;
  int mt = (r >> 1) & 15;   // 0..15 (channel tile of 16 in 256)
  int nt = r & 1;           // j tile
  int koff = (lane >> 4) << 1;
  int m = lane & 15;
  const float* Acol = Y1 + ((size_t)b * HW) * 768 + 512 + (mt * 16 + m);
  const float* Brow = L + ((size_t)(b * 32 + nt * 16 + m)) * HW;
  v8f acc = {0,0,0,0,0,0,0,0};
  for (int k0 = 0; k0 < HW; k0 += 4) {
    int h = k0 + koff;
    v2f a;
    a.x = Acol[(size_t)h * 768];
    a.y = Acol[(size_t)(h + 1) * 768];
    v2f bb = *(const v2f*)(Brow + h);
    acc = wmma4(a, bb, acc);
  }
  int j = nt * 16 + (lane & 15);
  if (j < 20) {
    int ch0 = mt * 16 + ((lane >> 4) << 3);
#pragma unroll
    for (int v = 0; v < 8; ++v) {
      int c = ch0 + v;
      float mu = murs[((size_t)b * NFAM + 384 + c) * 2];
      float rs = murs[((size_t)b * NFAM + 384 + c) * 2 + 1];
      outpre[((size_t)b * 256 + c) * 20 + j] = rs * acc[v] - rs * mu;
    }
  }
}

// ---------------------- lang4: out_l = W2 @ out_pre + b2  -> d_out lang region
__global__ __launch_bounds__(256) void lang4(const float* __restrict__ outpre,
                                             const float* __restrict__ W2,
                                             const float* __restrict__ b2,
                                             float* __restrict__ outl) {
  __shared__ float s_p[5120];
  int b = blockIdx.x;
  for (int i = threadIdx.x; i < 5120; i += 256) s_p[i] = outpre[(size_t)b * 5120 + i];
  __syncthreads();
  int c = threadIdx.x;  // 0..255
  float accj[20];
#pragma unroll
  for (int j = 0; j < 20; ++j) accj[j] = b2[c];
  for (int i = 0; i < 256; ++i) {
    float w = W2[c * 256 + i];
#pragma unroll
    for (int j = 0; j < 20; ++j) accj[j] = fmaf(w, s_p[i * 20 + j], accj[j]);
  }
#pragma unroll
  for (int j = 0; j < 20; ++j) outl[((size_t)b * 256 + c) * 20 + j] = accj[j];
}

// ---------------------------------------------------------------------------
extern "C" void kernel_launch(void* const* d_in, const int* in_sizes, int n_in,
                              void* d_out, int out_size, void* d_ws, size_t ws_size,
                              hipStream_t stream) {
  const float* x      = (const float*)d_in[0];
  const float* l      = (const float*)d_in[1];
  const float* lmask  = (const float*)d_in[2];
  const float* visW   = (const float*)d_in[3];
  const float* visB   = (const float*)d_in[4];
  const float* rvfqW  = (const float*)d_in[5];
  const float* rvfqB  = (const float*)d_in[6];
  const float* rvfq2W = (const float*)d_in[7];
  const float* rvfq2B = (const float*)d_in[8];
  const float* rvfkW  = (const float*)d_in[9];
  const float* rvfkB  = (const float*)d_in[10];
  const float* rlfqW  = (const float*)d_in[11];
  const float* rlfqB  = (const float*)d_in[12];
  const float* rlfq2W = (const float*)d_in[13];
  const float* rlfq2B = (const float*)d_in[14];
  const float* rlfkW  = (const float*)d_in[15];
  const float* rlfkB  = (const float*)d_in[16];
  const float* fvW    = (const float*)d_in[17];
  const float* fvB    = (const float*)d_in[18];
  const float* fvvW   = (const float*)d_in[19];
  const float* fvvB   = (const float*)d_in[20];
  const float* WW     = (const float*)d_in[21];
  const float* WB     = (const float*)d_in[22];
  const float* W2W    = (const float*)d_in[23];
  const float* W2B    = (const float*)d_in[24];
  const float* pmW    = (const float*)d_in[25];
  const float* pmB    = (const float*)d_in[26];
  // d_in[27]/[28]: softmax over a single scalar == 1.0 -> unused

  float* out = (float*)d_out;
  float* ws  = (float*)d_ws;
  float* Wbig   = ws + OFF_WBIG;
  float* bias   = ws + OFF_BIAS;
  float* sums   = ws + OFF_SUMS;
  float* murs   = ws + OFF_MURS;
  float* Wfold  = ws + OFF_WFOLD;
  float* bfold  = ws + OFF_BFOLD;
  float* kfold  = ws + OFF_KFOLD;
  float* Wv     = ws + OFF_WV;
  float* lbias  = ws + OFF_LBIAS;
  float* qlf    = ws + OFF_QLF;
  float* lconst = ws + OFF_LCONST;
  float* outpre = ws + OFF_OUTPRE;
  float* Y1     = ws + OFF_Y1;
  float* tq2    = ws + OFF_TQ2;
  float* tW     = ws + OFF_TW;
  float* L      = ws + OFF_L;
  float* outl   = out + (size_t)NB * HW * 128;

  zero_f<<<56, 256, 0, stream>>>(sums, NB * NFAM * 2);
  pack_w<<<388, 256, 0, stream>>>(visW, visB, rvfqW, rvfqB, rlfkW, rlfkB, fvvW, fvvB, Wbig, bias);
  k1_gemm<<<12288, 256, 0, stream>>>(x, Wbig, bias, Y1);
  stats_sum<<<dim3(64, 10, NB), 256, 0, stream>>>(Y1, 768, 128, 0, sums);      // q1,kl,vv
  stats_fin<<<20, 256, 0, stream>>>(sums, murs, 0, 640);
  k2b_wfold<<<NB, 128, 0, stream>>>(rvfq2W, rvfq2B, murs, Wfold, bfold);
  k3_gemm<<<2048, 256, 0, stream>>>(Y1, Wfold, bfold, tq2);
  stats_sum<<<dim3(64, 2, NB), 256, 0, stream>>>(tq2, 128, 0, 640, sums);      // q2
  stats_fin<<<4, 256, 0, stream>>>(sums, murs, 640, 128);
  small_pre<<<NB, 256, 0, stream>>>(l, lmask, rvfkW, rvfkB, fvW, fvB, WW,
                                    rlfqW, rlfqB, rlfq2W, rlfq2B, murs,
                                    kfold, Wv, lbias, qlf, lconst);
  k6_attn<<<NB * 64, 256, 0, stream>>>(tq2, kfold, Wv, lbias, WB, tW);
  stats_sum<<<dim3(64, 2, NB), 256, 0, stream>>>(tW, 128, 0, 768, sums);       // t_W
  stats_fin<<<4, 256, 0, stream>>>(sums, murs, 768, 128);
  k8_mm<<<2048, 256, 0, stream>>>(Y1, tW, murs, pmW, pmB, out);
  lang1<<<NB * 64, 256, 0, stream>>>(Y1, qlf, lconst, L);
  lang2<<<NB * 32, 256, 0, stream>>>(L);
  lang3_gemm<<<32, 256, 0, stream>>>(Y1, L, murs, outpre);
  lang4<<<NB, 256, 0, stream>>>(outpre, W2W, W2B, outl);
}